// SLayerExponential_43190191128605
// MI455X (gfx1250) — compile-verified
//
#include <hip/hip_runtime.h>

typedef __attribute__((ext_vector_type(2))) float v2f;
typedef __attribute__((ext_vector_type(8))) float v8f;

#define LOG2E_F 1.44269504088896340736f

// grid: (N/16, PSPLIT, B), block: 32 (one wave32)
__global__ __launch_bounds__(32)
void rbf_wmma_kernel(const float* __restrict__ batch,      // [B,P,2]
                     const float* __restrict__ not_dummy,  // [B,P]
                     const float* __restrict__ centers,    // [N,2]
                     const float* __restrict__ sharpness,  // [N,2]
                     float* __restrict__ out,              // [B,N]
                     int P, int N, int chunks_per_wave)
{
    const int lane   = threadIdx.x;          // 0..31
    const int ntile  = blockIdx.x;           // N/16 tiles
    const int psplit = blockIdx.y;
    const int b      = blockIdx.z;
    const bool hi    = lane >= 16;
    const int  l15   = lane & 15;

    // ---- Build B operand (per-center weights), constant for the wave ----
    // dist = w0*1 + w1*x0 + w2*x1 + w3*x0^2 + w4*x1^2, weights pre-scaled by
    // -log2(e). Feature K=5 is log2(nd) with weight 1.0, so the WMMA output is
    //   d = log2( nd * exp(-dist) )   ->  one v_exp_f32 per element, no mask mul.
    const int   n  = ntile * 16 + l15;
    const float c0 = centers[2 * n + 0], c1 = centers[2 * n + 1];
    const float s0 = sharpness[2 * n + 0], s1 = sharpness[2 * n + 1];
    const float k  = -LOG2E_F;
    const float w0 = (s0 * c0 * c0 + s1 * c1 * c1) * k;
    const float w1 = -2.0f * s0 * c0 * k;
    const float w2 = -2.0f * s1 * c1 * k;
    const float w3 = s0 * k;
    const float w4 = s1 * k;

    // B 4x16 f32 operand layout: VGPR0 = row K=0 (lanes 0-15) / K=2 (lanes 16-31),
    // VGPR1 = row K=1 / K=3. Second WMMA: rows K=4 (w4), K=5 (1.0), K=6/7 zero.
    v2f bm0, bm1;
    bm0.x = hi ? w2 : w0;
    bm0.y = hi ? w3 : w1;
    bm1.x = hi ? 0.0f : w4;
    bm1.y = hi ? 0.0f : 1.0f;

    const float* __restrict__ batch_b = batch + (size_t)b * P * 2;
    const float* __restrict__ nd_b    = not_dummy + (size_t)b * P;

    float acc0 = 0.0f, acc1 = 0.0f, acc2 = 0.0f, acc3 = 0.0f;
    const int pbase = psplit * chunks_per_wave * 16;

#pragma unroll 2
    for (int i = 0; i < chunks_per_wave; ++i) {
        const int p  = pbase + i * 16 + l15;
        const float x0 = batch_b[2 * p + 0];
        const float x1 = batch_b[2 * p + 1];
        const float lg = __builtin_amdgcn_logf(nd_b[p]);   // log2(nd); 0 -> -inf

        // A 16x4 f32 operand: lane<16 holds K=0,1 of row M=lane;
        // lane>=16 holds K=2,3 of row M=lane-16.
        // Features: (1, x0, x1, x0^2 | x1^2, log2(nd), 0, 0)
        v2f a0, a1;
        a0.x = hi ? x1 : 1.0f;
        a0.y = hi ? (x0 * x0) : x0;
        a1.x = hi ? 0.0f : (x1 * x1);
        a1.y = hi ? 0.0f : lg;

        v8f zero = {};
        // D = A1*B1 + (A0*B0 + 0): two chained v_wmma_f32_16x16x4_f32 (K=8 total)
        v8f d0 = __builtin_amdgcn_wmma_f32_16x16x4_f32(
            false, a0, false, bm0, (short)0, zero, false, false);
        v8f d = __builtin_amdgcn_wmma_f32_16x16x4_f32(
            false, a1, false, bm1, (short)0, d0, false, false);

        // D layout: lane L, reg r -> (M = r + 8*hi, N = L&15).
        // Element is log2(nd*exp(-dist)): a single exp2, then accumulate.
        acc0 += __builtin_amdgcn_exp2f(d[0]);
        acc1 += __builtin_amdgcn_exp2f(d[1]);
        acc2 += __builtin_amdgcn_exp2f(d[2]);
        acc3 += __builtin_amdgcn_exp2f(d[3]);
        acc0 += __builtin_amdgcn_exp2f(d[4]);
        acc1 += __builtin_amdgcn_exp2f(d[5]);
        acc2 += __builtin_amdgcn_exp2f(d[6]);
        acc3 += __builtin_amdgcn_exp2f(d[7]);
    }

    float acc = (acc0 + acc1) + (acc2 + acc3);

    // Columns n=L&15 are split across lanes L and L+16 (different M rows): fold.
    acc += __shfl_xor(acc, 16, 32);
    if (lane < 16) {
        atomicAdd(&out[(size_t)b * N + ntile * 16 + lane], acc);
    }
}

__global__ void zero_out_kernel(float* __restrict__ out, int nelem) {
    int i = blockIdx.x * blockDim.x + threadIdx.x;
    if (i < nelem) out[i] = 0.0f;
}

extern "C" void kernel_launch(void* const* d_in, const int* in_sizes, int n_in,
                              void* d_out, int out_size, void* d_ws, size_t ws_size,
                              hipStream_t stream) {
    const float* batch     = (const float*)d_in[0];  // [64,8192,2]
    const float* not_dummy = (const float*)d_in[1];  // [64,8192]
    const float* centers   = (const float*)d_in[2];  // [128,2]
    const float* sharpness = (const float*)d_in[3];  // [128,2]
    float* out = (float*)d_out;                      // [64,128]

    const int B = 64, P = 8192, N = 128;
    const int PSPLIT = 16;
    const int chunks_per_wave = P / (16 * PSPLIT);   // 32

    zero_out_kernel<<<(B * N + 255) / 256, 256, 0, stream>>>(out, B * N);

    dim3 grid(N / 16, PSPLIT, B);
    rbf_wmma_kernel<<<grid, 32, 0, stream>>>(batch, not_dummy, centers, sharpness,
                                             out, P, N, chunks_per_wave);
}